// ColBERTLikeScorer_77695958384860
// MI455X (gfx1250) — compile-verified
//
#include <hip/hip_runtime.h>
#include <hip/hip_bf16.h>
#include <math.h>

typedef __attribute__((ext_vector_type(2))) float v2f;
typedef __attribute__((ext_vector_type(8))) float v8f;

#define ALPHA_F 12.0f
#define INV_ALPHA 0.0833333333333333f
#define LOGIT_SCALE 14.285714285714286f
#define NB 16
#define LQ 128
#define DD 128
#define NO 128
#define LK 256
#define KS_STRIDE 132   // padded floats per LDS row -> conflict-free B loads

__global__ __launch_bounds__(256)
__attribute__((amdgpu_waves_per_eu(2)))
void colbert_score_kernel(const float* __restrict__ q,
                          const float* __restrict__ k,
                          const unsigned char* __restrict__ q_mask,
                          const unsigned char* __restrict__ k_mask,
                          float* __restrict__ out)
{
    extern __shared__ float smem[];
    float* Ks   = smem;                     // [LK][KS_STRIDE]
    float* wsum = smem + LK * KS_STRIDE;    // [8]

    const int o    = blockIdx.x;
    const int b    = blockIdx.y;
    const int tid  = threadIdx.x;
    const int wave = tid >> 5;
    const int lane = tid & 31;
    const int li   = lane & 15;   // column / row-in-tile index
    const int half = lane >> 4;   // K-half select per WMMA layout

    // ---- Stage K[o] into LDS, coalesced float4 copy ----
    const float4* k4 = (const float4*)(k + (size_t)o * LK * DD);
    for (int t = tid; t < LK * (DD / 4); t += 256) {
        int row = t >> 5;          // 32 float4 per row
        int c4  = t & 31;
        float4 v = k4[t];
        *(float4*)&Ks[row * KS_STRIDE + 4 * c4] = v;
    }
    __syncthreads();

    // ---- L2-normalize K rows in place (each wave: 32 rows) ----
    for (int r = wave * 32; r < wave * 32 + 32; ++r) {
        float4 x = *(float4*)&Ks[r * KS_STRIDE + 4 * lane];
        float ss = x.x * x.x + x.y * x.y + x.z * x.z + x.w * x.w;
        ss += __shfl_xor(ss, 1, 32);
        ss += __shfl_xor(ss, 2, 32);
        ss += __shfl_xor(ss, 4, 32);
        ss += __shfl_xor(ss, 8, 32);
        ss += __shfl_xor(ss, 16, 32);
        float inv = 1.0f / fmaxf(sqrtf(ss), 1e-12f);
        x.x *= inv; x.y *= inv; x.z *= inv; x.w *= inv;
        *(float4*)&Ks[r * KS_STRIDE + 4 * lane] = x;
    }
    __syncthreads();

    // ---- Load + normalize this wave's 16-row Q tile into registers ----
    // A-matrix (16x4 f32) layout: lane<16 holds M=li, K%4 in {0,1};
    //                             lane>=16 holds M=li, K%4 in {2,3}.
    const int i0 = wave * 16;
    const float* qrow = q + ((size_t)b * LQ + (i0 + li)) * DD;
    float a[64];
    float ssq = 0.f;
    #pragma unroll
    for (int s = 0; s < 32; ++s) {
        float2 v = *(const float2*)(qrow + 4 * s + 2 * half);
        a[2 * s] = v.x; a[2 * s + 1] = v.y;
        ssq += v.x * v.x + v.y * v.y;
    }
    ssq += __shfl_xor(ssq, 16, 32);           // join the two row halves
    float qinv = 1.0f / fmaxf(sqrtf(ssq), 1e-12f);
    #pragma unroll
    for (int s = 0; s < 64; ++s) a[s] *= qinv;

    // ---- hoist k_mask: bit jt = mask of this lane's column in tile jt ----
    const unsigned char* km_row = k_mask + (size_t)o * LK;
    unsigned mask_bits = 0;
    #pragma unroll
    for (int jt = 0; jt < 16; ++jt)
        mask_bits |= (km_row[jt * 16 + li] != 0) ? (1u << jt) : 0u;

    // ---- online logsumexp state: slot v = rows i0+v (half0), i0+v+8 (half1) ----
    float m[8], tsum[8];
    #pragma unroll
    for (int v = 0; v < 8; ++v) { m[v] = -INFINITY; tsum[v] = 0.f; }

    for (int jt = 0; jt < 16; ++jt) {
        const int j0 = jt * 16;
        const float* krow = &Ks[(j0 + li) * KS_STRIDE + 2 * half];
        v8f c = {0.f, 0.f, 0.f, 0.f, 0.f, 0.f, 0.f, 0.f};

        #pragma unroll
        for (int ch = 0; ch < 2; ++ch) {
            // batch-issue the chunk's 16 float2 B loads (8x ds_load_2addr_b64)
            float2 bv[16];
            #pragma unroll
            for (int p = 0; p < 16; ++p)
                bv[p] = *(const float2*)(krow + 4 * (16 * ch + p));
            // hard fence: loads may not sink below, WMMAs may not hoist above
            __builtin_amdgcn_sched_barrier(0);
            #pragma unroll
            for (int p = 0; p < 16; ++p) {
                const int s = 16 * ch + p;
                v2f av = { a[2 * s], a[2 * s + 1] };
                v2f bb = { bv[p].x, bv[p].y };
                c = __builtin_amdgcn_wmma_f32_16x16x4_f32(
                        false, av, false, bb, (short)0, c, false, false);
            }
        }

        const bool masked = (mask_bits >> jt) & 1u;
        #pragma unroll
        for (int v = 0; v < 8; ++v) {
            float z = masked ? -INFINITY : ALPHA_F * c[v];
            float rmax = z;
            rmax = fmaxf(rmax, __shfl_xor(rmax, 1, 32));
            rmax = fmaxf(rmax, __shfl_xor(rmax, 2, 32));
            rmax = fmaxf(rmax, __shfl_xor(rmax, 4, 32));
            rmax = fmaxf(rmax, __shfl_xor(rmax, 8, 32));
            float newm = fmaxf(m[v], rmax);
            float p = (newm == -INFINITY) ? 0.f : __expf(z - newm);
            p += __shfl_xor(p, 1, 32);
            p += __shfl_xor(p, 2, 32);
            p += __shfl_xor(p, 4, 32);
            p += __shfl_xor(p, 8, 32);
            float scale = (m[v] == -INFINITY) ? 0.f : __expf(m[v] - newm);
            tsum[v] = tsum[v] * scale + p;
            m[v] = newm;
        }
    }

    // ---- finalize rows: lse = (m + log t)/alpha, apply q_mask, sum 16 rows ----
    float acc = 0.f;
    #pragma unroll
    for (int v = 0; v < 8; ++v) {
        int row = i0 + v + 8 * half;
        float lse = (m[v] == -INFINITY) ? -INFINITY
                                        : (m[v] + __logf(tsum[v])) * INV_ALPHA;
        if (q_mask[(size_t)b * LQ + row]) lse = -INFINITY;
        acc += lse;
    }
    acc += __shfl_xor(acc, 16, 32);   // 16-row wave total (replicated)

    if (lane == 0) wsum[wave] = acc;
    __syncthreads();
    if (tid == 0) {
        float total = 0.f;
        #pragma unroll
        for (int w = 0; w < 8; ++w) total += wsum[w];
        if (!isfinite(total)) total = 0.f;
        total = total / (sqrtf((float)(LQ * LK)) + 1e-6f) * LOGIT_SCALE;
        out[(size_t)b * NO + o] = total;
    }
}

extern "C" void kernel_launch(void* const* d_in, const int* in_sizes, int n_in,
                              void* d_out, int out_size, void* d_ws, size_t ws_size,
                              hipStream_t stream) {
    (void)in_sizes; (void)n_in; (void)out_size; (void)d_ws; (void)ws_size;
    const float* q = (const float*)d_in[0];
    const float* k = (const float*)d_in[1];
    const unsigned char* qm = (const unsigned char*)d_in[2];
    const unsigned char* km = (const unsigned char*)d_in[3];
    float* out = (float*)d_out;

    const size_t smem = (size_t)LK * KS_STRIDE * sizeof(float) + 64;
    hipFuncSetAttribute((const void*)colbert_score_kernel,
                        hipFuncAttributeMaxDynamicSharedMemorySize, (int)smem);

    colbert_score_kernel<<<dim3(NO, NB, 1), 256, smem, stream>>>(q, k, qm, km, out);
}